// RecModel_86131274154603
// MI455X (gfx1250) — compile-verified
//
#include <hip/hip_runtime.h>

// ---------------------------------------------------------------------------
// RecModel scan, MI455X (gfx1250).
// x_{t+1} = M_t x_t + c_t ;  M_t = I + (A - xic_t C) dt ; c_t = xic_t dy_t
// out_t   = (C x_t) dt  (previous state)
//
// 3-pass blocked affine scan (compose chunk maps -> scan maps -> replay/emit).
// Inputs streamed HBM -> LDS with CDNA5 async data-mover ops
// (global_load_async_to_lds_*, s_wait_asynccnt), double-buffered in private
// per-thread LDS slabs (no barriers needed). The async INST_OFFSET applies to
// BOTH the LDS and global address (ISA 08 §4.4), so step offsets are encoded
// as immediates and base pointers advance once per round.
// Slab strides padded (272B / 136B) so ds_load lane stride = 68 / 34 dwords
// (== 4 / 34 mod 64 banks) -> conflict-free LDS reads.
// ---------------------------------------------------------------------------

namespace {
constexpr int   kB      = 512;
constexpr int   kT      = 8192;
constexpr int   kL      = 64;            // chunk length (timesteps)
constexpr int   kG      = kT / kL;       // 128 chunks per batch element
constexpr int   kSub    = 8;             // timesteps staged per round
constexpr int   kRounds = kL / kSub;     // 8 staging rounds per chunk
constexpr int   kTPB    = 128;           // threads per block (4 waves)
constexpr float kDT     = 1e-3f;
constexpr unsigned kXSlab = 2 * kSub * 16 + 16;  // 272 B per thread (padded)
constexpr unsigned kDSlab = 2 * kSub * 8  + 8;   // 136 B per thread (padded)
static_assert(kG == 128, "shift tricks assume 128 chunks per batch");
static_assert((kB * kG) % kTPB == 0, "grid must tile exactly");
}

// ---- CDNA5 async global->LDS copies (tracked by ASYNCcnt) ------------------
// wait until <= N async ops outstanding (loads signal done oldest-first)
__device__ __forceinline__ void wait_async_16() {
  asm volatile("s_wait_asynccnt 0x10" ::: "memory");
}
__device__ __forceinline__ void wait_async_0() {
  asm volatile("s_wait_asynccnt 0x0" ::: "memory");
}

// low 32 bits of a flat pointer to a __shared__ object == LDS byte address
__device__ __forceinline__ unsigned lds_addr_of(const void* p) {
  return (unsigned)(unsigned long long)(size_t)p;
}

// Issue one staging round: kSub x (16B xic + 8B dy) copies with compile-time
// INST_OFFSETs (added to both LDS and global addresses by hardware).
template <int K>
__device__ __forceinline__ void issue_steps(unsigned xl, unsigned dl,
                                            const char* xg, const char* dg) {
  if constexpr (K < kSub) {
    asm volatile("global_load_async_to_lds_b128 %0, %1, off offset:%2"
                 :: "v"(xl), "v"(xg), "n"(K * 16) : "memory");
    asm volatile("global_load_async_to_lds_b64 %0, %1, off offset:%2"
                 :: "v"(dl), "v"(dg), "n"(K * 8) : "memory");
    issue_steps<K + 1>(xl, dl, xg, dg);
  }
}

// ---------------------------------------------------------------------------
// Pass 1: per-chunk affine composition.  thread n -> (b = n/128, g = n%128)
// P starts as I, q as 0;  per step: P <- M P ; q <- M q + c.
// ---------------------------------------------------------------------------
__global__ __launch_bounds__(kTPB) void rec_compose(
    const float* __restrict__ xic, const float* __restrict__ dy,
    const float* __restrict__ A,   const float* __restrict__ C,
    float4* __restrict__ Pout, float2* __restrict__ qout)
{
  __shared__ __align__(16) char Xraw[kTPB * kXSlab];   // ~34 KB
  __shared__ __align__(16) char Draw[kTPB * kDSlab];   // ~17 KB

  const int tid = threadIdx.x;
  const int n   = blockIdx.x * kTPB + tid;
  const size_t t0 = ((size_t)(n >> 7) * kT) + (size_t)(n & (kG - 1)) * kL;
  const char* xg = (const char*)xic + t0 * 16;  // this lane's xic stream
  const char* dg = (const char*)dy  + t0 * 8;   // this lane's dy stream

  const float a00 = A[0], a01 = A[1], a10 = A[2], a11 = A[3];
  const float c00 = C[0], c01 = C[1], c10 = C[2], c11 = C[3];

  const unsigned xslab = lds_addr_of(Xraw) + (unsigned)tid * kXSlab;
  const unsigned dslab = lds_addr_of(Draw) + (unsigned)tid * kDSlab;

  float P00 = 1.f, P01 = 0.f, P10 = 0.f, P11 = 1.f, q0 = 0.f, q1 = 0.f;

  issue_steps<0>(xslab, dslab, xg, dg);
  for (int s = 0; s < kRounds; ++s) {
    const int buf = s & 1;
    if (s + 1 < kRounds) {
      xg += kSub * 16; dg += kSub * 8;
      issue_steps<0>(xslab + (unsigned)(buf ^ 1) * (kSub * 16),
                     dslab + (unsigned)(buf ^ 1) * (kSub * 8), xg, dg);
      wait_async_16();           // oldest 16 (this round's copies) are done
    } else {
      wait_async_0();
    }
    const float4* Xp = (const float4*)(Xraw + (size_t)tid * kXSlab + buf * (kSub * 16));
    const float2* Dp = (const float2*)(Draw + (size_t)tid * kDSlab + buf * (kSub * 8));
#pragma unroll
    for (int k = 0; k < kSub; ++k) {
      const float4 X = Xp[k];    // xic[0][0..1], xic[1][0..1]
      const float2 D = Dp[k];
      const float xc00 = X.x * c00 + X.y * c10;
      const float xc01 = X.x * c01 + X.y * c11;
      const float xc10 = X.z * c00 + X.w * c10;
      const float xc11 = X.z * c01 + X.w * c11;
      const float m00 = fmaf(a00 - xc00, kDT, 1.0f);
      const float m01 = (a01 - xc01) * kDT;
      const float m10 = (a10 - xc10) * kDT;
      const float m11 = fmaf(a11 - xc11, kDT, 1.0f);
      const float cv0 = X.x * D.x + X.y * D.y;
      const float cv1 = X.z * D.x + X.w * D.y;
      const float p00 = m00 * P00 + m01 * P10;
      const float p01 = m00 * P01 + m01 * P11;
      const float p10 = m10 * P00 + m11 * P10;
      const float p11 = m10 * P01 + m11 * P11;
      const float nq0 = m00 * q0 + m01 * q1 + cv0;
      const float nq1 = m10 * q0 + m11 * q1 + cv1;
      P00 = p00; P01 = p01; P10 = p10; P11 = p11; q0 = nq0; q1 = nq1;
    }
  }

  Pout[n] = make_float4(P00, P01, P10, P11);
  qout[n] = make_float2(q0, q1);
}

// ---------------------------------------------------------------------------
// Pass 2: serial scan over chunk maps, one lane per batch element.
// Stores x at each chunk *start*.  1.5 MB through L2 -> negligible.
// ---------------------------------------------------------------------------
__global__ __launch_bounds__(256) void rec_scan(
    const float4* __restrict__ P, const float2* __restrict__ q,
    float2* __restrict__ st)
{
  const int b = blockIdx.x * blockDim.x + threadIdx.x;
  if (b >= kB) return;
  float x0 = 1.0f, x1 = 0.0f;   // initial_state = (1, 0)
  for (int g = 0; g < kG; ++g) {
    const int idx = b * kG + g;
    st[idx] = make_float2(x0, x1);
    const float4 Pv = P[idx];
    const float2 qv = q[idx];
    const float nx0 = Pv.x * x0 + Pv.y * x1 + qv.x;
    const float nx1 = Pv.z * x0 + Pv.w * x1 + qv.y;
    x0 = nx0; x1 = nx1;
  }
}

// ---------------------------------------------------------------------------
// Pass 3: replay each chunk from its start state, emit out_t = C x_{t-1} dt.
// Uses dx = A x dt + X (dy - out)  [since (A - XC)x dt + X dy
//                                   = A x dt - X (C x dt) + X dy] -> no XC.
// ---------------------------------------------------------------------------
__global__ __launch_bounds__(kTPB) void rec_emit(
    const float* __restrict__ xic, const float* __restrict__ dy,
    const float* __restrict__ A,   const float* __restrict__ C,
    const float2* __restrict__ st, float2* __restrict__ out)
{
  __shared__ __align__(16) char Xraw[kTPB * kXSlab];
  __shared__ __align__(16) char Draw[kTPB * kDSlab];

  const int tid = threadIdx.x;
  const int n   = blockIdx.x * kTPB + tid;
  const size_t t0 = ((size_t)(n >> 7) * kT) + (size_t)(n & (kG - 1)) * kL;
  const char* xg = (const char*)xic + t0 * 16;
  const char* dg = (const char*)dy  + t0 * 8;
  float2* op = out + t0;

  const float a00 = A[0], a01 = A[1], a10 = A[2], a11 = A[3];
  const float c00 = C[0], c01 = C[1], c10 = C[2], c11 = C[3];

  const unsigned xslab = lds_addr_of(Xraw) + (unsigned)tid * kXSlab;
  const unsigned dslab = lds_addr_of(Draw) + (unsigned)tid * kDSlab;

  const float2 x_init = st[n];
  float x0 = x_init.x, x1 = x_init.y;

  issue_steps<0>(xslab, dslab, xg, dg);
  for (int s = 0; s < kRounds; ++s) {
    const int buf = s & 1;
    if (s + 1 < kRounds) {
      xg += kSub * 16; dg += kSub * 8;
      issue_steps<0>(xslab + (unsigned)(buf ^ 1) * (kSub * 16),
                     dslab + (unsigned)(buf ^ 1) * (kSub * 8), xg, dg);
      wait_async_16();
    } else {
      wait_async_0();
    }
    const float4* Xp = (const float4*)(Xraw + (size_t)tid * kXSlab + buf * (kSub * 16));
    const float2* Dp = (const float2*)(Draw + (size_t)tid * kDSlab + buf * (kSub * 8));
#pragma unroll
    for (int k = 0; k < kSub; ++k) {
      const float4 X = Xp[k];
      const float2 D = Dp[k];
      // output uses previous state
      const float o0 = (c00 * x0 + c01 * x1) * kDT;
      const float o1 = (c10 * x0 + c11 * x1) * kDT;
      op[s * kSub + k] = make_float2(o0, o1);
      const float e0 = D.x - o0;
      const float e1 = D.y - o1;
      const float ax0 = a00 * x0 + a01 * x1;
      const float ax1 = a10 * x0 + a11 * x1;
      x0 += fmaf(ax0, kDT, X.x * e0 + X.y * e1);
      x1 += fmaf(ax1, kDT, X.z * e0 + X.w * e1);
    }
  }
}

// ---------------------------------------------------------------------------
extern "C" void kernel_launch(void* const* d_in, const int* in_sizes, int n_in,
                              void* d_out, int out_size, void* d_ws, size_t ws_size,
                              hipStream_t stream) {
  (void)in_sizes; (void)n_in; (void)out_size; (void)ws_size;
  const float* xic = (const float*)d_in[0];   // [B,T,2,2]
  const float* dy  = (const float*)d_in[1];   // [B,T,2]
  const float* A   = (const float*)d_in[2];   // [2,2]
  const float* C   = (const float*)d_in[3];   // [2,2]

  // workspace layout: P (1 MB) | q (0.5 MB) | chunk-start states (0.5 MB)
  float4* Pw = (float4*)d_ws;
  float2* qw = (float2*)(Pw + (size_t)kB * kG);
  float2* st = qw + (size_t)kB * kG;

  const dim3 grid((kB * kG) / kTPB);   // 512 blocks x 128 threads
  rec_compose<<<grid, kTPB, 0, stream>>>(xic, dy, A, C, Pw, qw);
  rec_scan<<<(kB + 255) / 256, 256, 0, stream>>>(Pw, qw, st);
  rec_emit<<<grid, kTPB, 0, stream>>>(xic, dy, A, C, st, (float2*)d_out);
}